// CRFDecoder_14010183320053
// MI455X (gfx1250) — compile-verified
//
#include <hip/hip_runtime.h>

// Problem constants (from reference): B=32, T=2048, D=512, K=16
constexpr int Bn = 32;
constexpr int Tn = 2048;
constexpr int Dn = 512;
constexpr int Kn = 16;

typedef float v2f __attribute__((ext_vector_type(2)));
typedef float v4f __attribute__((ext_vector_type(4)));
typedef float v8f __attribute__((ext_vector_type(8)));
typedef int   v4i __attribute__((ext_vector_type(4)));

#define AS1 __attribute__((address_space(1)))
#define AS3 __attribute__((address_space(3)))

#if __has_builtin(__builtin_amdgcn_global_load_async_to_lds_b128)
#  define HAVE_ASYNC_LDS 1
#else
#  define HAVE_ASYNC_LDS 0
#endif

// ---------------------------------------------------------------------------
// Kernel 1: linear_logits[m, k] = sum_d X[m, d] * W[k, d] + bias[k]
//   M = B*T = 65536 rows, N = K = 16, reduce-K = D = 512.
//   4 waves / block, each wave owns one 16-row M tile; K chunked by 64.
//   A tiles staged memory->LDS with GLOBAL_LOAD_ASYNC_TO_LDS_B128 (ASYNCcnt).
// ---------------------------------------------------------------------------
constexpr int WAVES_PER_BLOCK = 4;
constexpr int ROWS_PER_BLOCK  = 16 * WAVES_PER_BLOCK;   // 64
constexpr int CW              = 64;                      // K-chunk (floats)
constexpr int ASTRIDE         = CW + 4;                  // 68: conflict-free b64 reads

__global__ __launch_bounds__(32 * WAVES_PER_BLOCK)
void crf_gemm_wmma(const float* __restrict__ X,      // [M, D]
                   const float* __restrict__ W,      // [K, D]
                   const float* __restrict__ bias,   // [K]
                   float* __restrict__ Y)            // [M, K]
{
    // B fragments: for k-step k4, lane L holds {B[4k4+2h][n], B[4k4+2h+1][n]}
    // with n = L&15, h = L>>4, and B[d][n] = W[n][d].
    __shared__ float bfrag[Dn / 4][32][2];                       // 32 KB
    __shared__ float atile[WAVES_PER_BLOCK][16 * ASTRIDE];       // 17 KB

    const int tid  = threadIdx.x;
    const int lane = tid & 31;
    const int wave = tid >> 5;

    // Stage W^T fragments once per block (W is 32 KB, stays in L2).
    for (int idx = tid; idx < (Dn / 4) * 32; idx += blockDim.x) {
        const int k4 = idx >> 5;
        const int l  = idx & 31;
        const int n  = l & 15;
        const int d0 = 4 * k4 + 2 * (l >> 4);
        const float* wp = W + n * Dn + d0;
        bfrag[k4][l][0] = wp[0];
        bfrag[k4][l][1] = wp[1];
    }
    __syncthreads();

    const int    rowBase = blockIdx.x * ROWS_PER_BLOCK + wave * 16;
    const float* Xw      = X + (size_t)rowBase * Dn;
    float*       aw      = atile[wave];

    const int arow = lane & 15;
    const int koff = 2 * (lane >> 4);

    v8f acc = {};
    for (int c = 0; c < Dn / CW; ++c) {
        // WAR: previous chunk's fragment ds_loads must drain before the async
        // unit overwrites atile (async LDS writes are NOT ordered with DS ops).
        asm volatile("s_wait_dscnt 0" ::: "memory");

        // Stage 16 rows x 64 cols (4 KB) into LDS, fully coalesced.
        #pragma unroll
        for (int it = 0; it < 8; ++it) {
            const int idx  = it * 32 + lane;     // 256 float4 elements
            const int r    = idx >> 4;           // 16 float4 per row
            const int colq = idx & 15;
#if HAVE_ASYNC_LDS
            __builtin_amdgcn_global_load_async_to_lds_b128(
                (AS1 v4i*)(Xw + r * Dn + c * CW + colq * 4),
                (AS3 v4i*)(aw + r * ASTRIDE + colq * 4), 0, 0);
#else
            v4f v = *(const v4f*)(Xw + r * Dn + c * CW + colq * 4);
            *(v4f*)(aw + r * ASTRIDE + colq * 4) = v;
#endif
        }
#if HAVE_ASYNC_LDS
        asm volatile("s_wait_asynccnt 0" ::: "memory");   // RAW: tile landed in LDS
#else
        asm volatile("s_wait_dscnt 0" ::: "memory");      // same-wave DS in-order
#endif

        #pragma unroll
        for (int k4 = 0; k4 < CW / 4; ++k4) {
            v2f a = *(const v2f*)(aw + arow * ASTRIDE + k4 * 4 + koff);
            v2f b = *(const v2f*)(&bfrag[c * (CW / 4) + k4][lane][0]);
            acc = __builtin_amdgcn_wmma_f32_16x16x4_f32(
                false, a, false, b, (short)0, acc, false, false);
        }
        asm volatile("" ::: "memory");   // keep next chunk's staging after reads
    }

    // C/D layout: VGPR v, lanes 0-15 -> row v, lanes 16-31 -> row v+8; col = lane&15.
    const float bv   = bias[lane & 15];
    const int   mrow = 8 * (lane >> 4);
    float* yp = Y + (size_t)(rowBase + mrow) * Kn + (lane & 15);
    #pragma unroll
    for (int v = 0; v < 8; ++v)
        yp[v * Kn] = acc[v] + bv;
}

// ---------------------------------------------------------------------------
// Kernel 2: Viterbi forward + backtrace + one-hot, one wave per batch.
//   Lane L: state j = L&15, half h = L>>4 reduces prev-states h*8..h*8+7.
// ---------------------------------------------------------------------------
__global__ __launch_bounds__(32)
void crf_viterbi(const float* __restrict__ E,            // [B,T,K] linear logits
                 const unsigned char* __restrict__ mask, // [B,T]
                 const float* __restrict__ trans,        // [K,K]
                 const float* __restrict__ start_t,      // [K]
                 const float* __restrict__ end_t,        // [K]
                 float* __restrict__ crf)                // [B,T,K]
{
    __shared__ unsigned char ptrs[Tn * Kn];              // 32 KB backpointers

    const int b    = blockIdx.x;
    const int lane = threadIdx.x;
    const int j    = lane & 15;
    const int h    = lane >> 4;

    const float*         Eb   = E    + (size_t)b * Tn * Kn;
    const unsigned char* mb   = mask + (size_t)b * Tn;
    float*               crfb = crf  + (size_t)b * Tn * Kn;

    // Per-lane transition column slice: tc[ii] = trans[h*8+ii][j]
    float tc[8];
    #pragma unroll
    for (int ii = 0; ii < 8; ++ii)
        tc[ii] = trans[(h * 8 + ii) * Kn + j];

    // init: score[j] = start_t[j] + emissions[b, 0, j]  (replicated in both halves)
    float s = start_t[j] + Eb[j];

    float emit_next = Eb[Kn + j];
    int   m_next    = mb[1];

    for (int t = 1; t < Tn; ++t) {
        const float emit = emit_next;
        const int   mcur = m_next;
        if (t + 1 < Tn) {
            emit_next = Eb[(t + 1) * Kn + j];
            m_next    = mb[t + 1];
        }
        if (t + 12 < Tn)
            __builtin_prefetch(Eb + (t + 12) * Kn, 0, 0);

        // half-reduction: max_i (s[i] + trans[i][j]) over i in [8h, 8h+8)
        float best = -3.402823466e38f;
        int   bi   = 0;
        #pragma unroll
        for (int ii = 0; ii < 8; ++ii) {
            const float si   = __shfl(s, h * 8 + ii, 32);
            const float cand = si + tc[ii];
            const bool  gt   = cand > best;        // strict '>' => first index wins
            bi   = gt ? (h * 8 + ii) : bi;
            best = gt ? cand : best;
        }
        // combine the two halves (prefer lower index block on ties)
        const float obest = __shfl_xor(best, 16, 32);
        const int   obi   = __shfl_xor(bi, 16, 32);
        const bool  takeOther = h ? (obest >= best) : (obest > best);
        best = takeOther ? obest : best;
        bi   = takeOther ? obi   : bi;

        const float ns = mcur ? (best + emit) : s;
        const int   np = mcur ? bi : j;
        if (h == 0)
            ptrs[t * Kn + j] = (unsigned char)np;
        s = ns;
    }

    // final tag = argmax_j (score[j] + end_t[j]), first index on ties
    const float fs = s + end_t[j];
    float best = -3.402823466e38f;
    int   tag  = 0;
    for (int i = 0; i < 16; ++i) {
        const float v = __shfl(fs, i, 32);
        if (v > best) { best = v; tag = i; }
    }

    asm volatile("s_wait_dscnt 0" ::: "memory");   // ptr stores ordered before loads

    // backtrace + one-hot (crf[t, j] = (j == tag) * mask[t])
    for (int t = Tn - 1; t >= 1; --t) {
        if (h == 0)
            crfb[t * Kn + j] = ((j == tag) && mb[t]) ? 1.0f : 0.0f;
        tag = (int)ptrs[t * Kn + tag];             // uniform dependent LDS chain
    }
    if (h == 0)
        crfb[j] = ((j == tag) && mb[0]) ? 1.0f : 0.0f;
}

// ---------------------------------------------------------------------------
extern "C" void kernel_launch(void* const* d_in, const int* in_sizes, int n_in,
                              void* d_out, int out_size, void* d_ws, size_t ws_size,
                              hipStream_t stream) {
    const float*         logits  = (const float*)d_in[0];         // [B,T,D]
    const unsigned char* mask    = (const unsigned char*)d_in[1]; // [B,T] bool
    const float*         W       = (const float*)d_in[2];         // [K,D]
    const float*         bias    = (const float*)d_in[3];         // [K]
    const float*         trans   = (const float*)d_in[4];         // [K,K]
    const float*         start_t = (const float*)d_in[5];         // [K]
    const float*         end_t   = (const float*)d_in[6];         // [K]

    float* lin = (float*)d_out;                   // [B,T,K]
    float* crf = lin + (size_t)Bn * Tn * Kn;      // [B,T,K]

    crf_gemm_wmma<<<(Bn * Tn) / ROWS_PER_BLOCK, 32 * WAVES_PER_BLOCK, 0, stream>>>(
        logits, W, bias, lin);
    crf_viterbi<<<Bn, 32, 0, stream>>>(lin, mask, trans, start_t, end_t, crf);
}